// SSIMLoss_86157043957784
// MI455X (gfx1250) — compile-verified
//
#include <hip/hip_runtime.h>

typedef float v8f __attribute__((ext_vector_type(8)));
typedef float v2f __attribute__((ext_vector_type(2)));

#define WIN   11
#define RAD   5
#define C1F   1e-4f          // 0.01^2
#define C2F   9e-4f          // 0.03^2

#define IMG_H 384
#define IMG_W 640
#define NPLANE 96            // 32 batch * 3 channels
#define RY    32             // output rows per block
#define RX    64             // output cols per block
#define HR    (RY + WIN - 1) // 42 rows of horizontal sums
#define HSTR  (RX + 1)       // 65: LDS stride, bank = (r + c) mod 64

// ---------------------------------------------------------------------------
// Kernel 1: fused SSIM. Per block: 32x64 output region.
//   Stage A (VALU): horizontal 11-tap sliding sums of {p,t,pp,tt,pt} -> LDS.
//   Stage B (WMMA): vertical 11-tap sums as band-matrix x H via
//                   v_wmma_f32_16x16x4_f32, 7 K-chunks x 5 quantities / tile.
//   Stage C: per-lane SSIM + deterministic LDS tree reduction -> partial[].
// ---------------------------------------------------------------------------
__global__ __launch_bounds__(256)
void ssim_main(const float* __restrict__ pred,
               const float* __restrict__ target,
               float* __restrict__ partial) {
    __shared__ float Hs[5][HR * HSTR];   // 54,600 bytes

    const int tid = threadIdx.x;
    const int bx  = blockIdx.x;
    const int nrx = IMG_W / RX;          // 10
    const int nry = IMG_H / RY;          // 12
    const int rx    = bx % nrx;
    const int ry    = (bx / nrx) % nry;
    const int plane = bx / (nrx * nry);
    const int x0 = rx * RX;
    const int y0 = ry * RY;
    const int base = plane * (IMG_H * IMG_W);

    // ---------------- Stage A: horizontal box sums ----------------
    // 168 tasks: (row r in [0,42), 16-col chunk c in [0,4))
    if (tid < HR * 4) {
        const int r      = tid >> 2;
        const int cchunk = tid & 3;
        const int gy     = y0 + r - RAD;
        const bool rowOK = (gy >= 0) && (gy < IMG_H);
        const float* prow = pred   + base + gy * IMG_W;
        const float* trow = target + base + gy * IMG_W;
        const int xbase = x0 + cchunk * 16;      // first output col (absolute)

        float sp = 0.f, st = 0.f, spp = 0.f, stt = 0.f, spt = 0.f;
        #pragma unroll
        for (int d = -RAD; d <= RAD; ++d) {      // prime the window
            const int gx = xbase + d;
            float p = 0.f, t = 0.f;
            if (rowOK && gx >= 0 && gx < IMG_W) { p = prow[gx]; t = trow[gx]; }
            sp += p; st += t; spp += p * p; stt += t * t; spt += p * t;
        }
        const int hb = r * HSTR + cchunk * 16;
        Hs[0][hb] = sp; Hs[1][hb] = st; Hs[2][hb] = spp;
        Hs[3][hb] = stt; Hs[4][hb] = spt;

        for (int i = 1; i < 16; ++i) {           // slide: +new -old
            const int gxN = xbase + i + RAD;
            const int gxO = xbase + i - RAD - 1;
            float pn = 0.f, tn = 0.f, po = 0.f, to = 0.f;
            if (rowOK && gxN < IMG_W)            { pn = prow[gxN]; tn = trow[gxN]; }
            if (rowOK && gxO >= 0 && gxO < IMG_W){ po = prow[gxO]; to = trow[gxO]; }
            sp  += pn - po;            st  += tn - to;
            spp += pn * pn - po * po;  stt += tn * tn - to * to;
            spt += pn * tn - po * to;
            Hs[0][hb + i] = sp; Hs[1][hb + i] = st; Hs[2][hb + i] = spp;
            Hs[3][hb + i] = stt; Hs[4][hb + i] = spt;
        }
    }
    __syncthreads();

    // ---------------- Stage B: vertical box sums via f32 WMMA -------------
    // Wave w owns tile (ty = w>>2, tx = w&3) of 16x16 outputs.
    const int wave = tid >> 5;
    const int lane = tid & 31;
    const int n    = lane & 15;      // N / M index within half-wave
    const int lh   = lane >> 4;      // half-wave selector
    const int ty   = wave >> 2;      // 0..1
    const int tx   = wave & 3;       // 0..3

    v8f acc[5];
    #pragma unroll
    for (int q = 0; q < 5; ++q)
        #pragma unroll
        for (int j = 0; j < 8; ++j) acc[q][j] = 0.f;

    const int colOff = tx * 16 + n;
    #pragma unroll
    for (int k0 = 0; k0 < 28; k0 += 4) {
        // A = 16x26 band matrix chunk (ones where 0 <= r_band - y <= 10),
        // generated in registers; shared across all 5 quantities.
        v2f a;
        const int r0 = k0 + 2 * lh;              // band row for A vgpr 0
        a[0] = ((unsigned)(r0     - n) <= 10u) ? 1.f : 0.f;
        a[1] = ((unsigned)(r0 + 1 - n) <= 10u) ? 1.f : 0.f;

        const int hr0 = ty * 16 + k0 + 2 * lh;   // H row for B vgpr 0
        const int r0c = min(hr0,     HR - 1);    // clamped rows have zero A
        const int r1c = min(hr0 + 1, HR - 1);

        #pragma unroll
        for (int q = 0; q < 5; ++q) {
            v2f b;
            b[0] = Hs[q][r0c * HSTR + colOff];
            b[1] = Hs[q][r1c * HSTR + colOff];
            acc[q] = __builtin_amdgcn_wmma_f32_16x16x4_f32(
                false, a, false, b, (short)0, acc[q], false, false);
        }
    }

    // ---------------- Stage C: SSIM per pixel + reduction ----------------
    const float inv = 1.0f / 121.0f;
    float lsum = 0.f;
    #pragma unroll
    for (int v = 0; v < 8; ++v) {
        const float mu_p = acc[0][v] * inv;
        const float mu_t = acc[1][v] * inv;
        const float e_pp = acc[2][v] * inv;
        const float e_tt = acc[3][v] * inv;
        const float e_pt = acc[4][v] * inv;
        const float sig_p  = e_pp - mu_p * mu_p;
        const float sig_t  = e_tt - mu_t * mu_t;
        const float sig_pt = e_pt - mu_p * mu_t;
        const float num = (2.f * mu_p * mu_t + C1F) * (2.f * sig_pt + C2F);
        const float den = (mu_p * mu_p + mu_t * mu_t + C1F) *
                          (sig_p + sig_t + C2F);
        lsum += num / den;
    }

    __syncthreads();                 // Hs no longer needed; reuse as scratch
    float* red = (float*)Hs;
    red[tid] = lsum;
    __syncthreads();
    #pragma unroll
    for (int s = 128; s > 0; s >>= 1) {
        if (tid < s) red[tid] += red[tid + s];
        __syncthreads();
    }
    if (tid == 0) partial[bx] = red[0];
}

// ---------------------------------------------------------------------------
// Kernel 2: deterministic final reduction (fixed order, no float atomics).
// ---------------------------------------------------------------------------
__global__ __launch_bounds__(256)
void ssim_reduce(const float* __restrict__ partial, int nblocks,
                 float* __restrict__ out) {
    __shared__ float red[256];
    float s = 0.f;
    for (int i = threadIdx.x; i < nblocks; i += 256) s += partial[i];
    red[threadIdx.x] = s;
    __syncthreads();
    #pragma unroll
    for (int k = 128; k > 0; k >>= 1) {
        if (threadIdx.x < k) red[threadIdx.x] += red[threadIdx.x + k];
        __syncthreads();
    }
    if (threadIdx.x == 0)
        out[0] = 1.0f - red[0] / (float)(NPLANE * IMG_H * IMG_W);
}

extern "C" void kernel_launch(void* const* d_in, const int* in_sizes, int n_in,
                              void* d_out, int out_size, void* d_ws, size_t ws_size,
                              hipStream_t stream) {
    (void)in_sizes; (void)n_in; (void)out_size; (void)ws_size;
    const float* pred   = (const float*)d_in[0];
    const float* target = (const float*)d_in[1];
    float* out = (float*)d_out;
    float* ws  = (float*)d_ws;   // per-block partial sums (11520 floats)

    const int nblocks = NPLANE * (IMG_H / RY) * (IMG_W / RX);   // 11520
    ssim_main<<<nblocks, 256, 0, stream>>>(pred, target, ws);
    ssim_reduce<<<1, 256, 0, stream>>>(ws, nblocks, out);
}